// GMNLayer_6433861009688
// MI455X (gfx1250) — compile-verified
//
#include <hip/hip_runtime.h>
#include <hip/hip_bf16.h>
#include <math.h>

typedef __attribute__((ext_vector_type(2))) float v2f;
typedef __attribute__((ext_vector_type(8))) float v8f;

#define Bq 4
#define Nn 24
#define Ee 72
#define Dd 512
#define TWO_D 1024
#define NE (Bq * Ee)          // 288 rows of X/Y/T
#define RR (Nn * Nn)          // 576, power-iteration vector length
#define PI_ITERS 50
#define SH_ITERS 20

// ---------------------------------------------------------------------------
// Kernel A: W = relu of [[l1,l2],[l2,l1]] with lX = relu(lambdaX + lambdaX^T)
// ---------------------------------------------------------------------------
__global__ void build_w(const float* __restrict__ l1, const float* __restrict__ l2,
                        float* __restrict__ W) {
    int idx = blockIdx.x * blockDim.x + threadIdx.x;   // < 1024*1024
    int p = idx >> 10, q = idx & 1023;
    int a = p & 511, c = q & 511;
    const float* L = ((p < 512) == (q < 512)) ? l1 : l2;
    float v = L[a * 512 + c] + L[c * 512 + a];
    W[idx] = fmaxf(v, 0.0f);
}

// ---------------------------------------------------------------------------
// Kernel B: gather per-edge endpoint features (one-hot incidence -> row pick)
// X[b,e,:] = [F_src[gsrc(e)], F_src[hsrc(e)]]; Y likewise with tgt.
// ---------------------------------------------------------------------------
__global__ void gather_xy(const float* __restrict__ F_src, const float* __restrict__ F_tgt,
                          const float* __restrict__ G_src, const float* __restrict__ H_src,
                          const float* __restrict__ G_tgt, const float* __restrict__ H_tgt,
                          float* __restrict__ X, float* __restrict__ Y) {
    int blk = blockIdx.x;                    // 0..575
    int isY = (blk >= NE);
    int be  = isY ? blk - NE : blk;
    int b = be / Ee, e = be % Ee;
    const float* G = isY ? G_tgt : G_src;
    const float* H = isY ? H_tgt : H_src;
    const float* F = isY ? F_tgt : F_src;
    float* O = isY ? Y : X;
    const float* gcol = G + b * Nn * Ee + e;
    const float* hcol = H + b * Nn * Ee + e;
    int gi = 0, hi = 0;
    for (int n = 0; n < Nn; ++n) {
        if (gcol[n * Ee] > 0.5f) gi = n;
        if (hcol[n * Ee] > 0.5f) hi = n;
    }
    const float* fg = F + (b * Nn + gi) * Dd;
    const float* fh = F + (b * Nn + hi) * Dd;
    float* orow = O + be * TWO_D;
    for (int d = threadIdx.x; d < Dd; d += blockDim.x) {
        orow[d]      = fg[d];
        orow[Dd + d] = fh[d];
    }
}

// ---------------------------------------------------------------------------
// Kernel C: T = X * W   ([288,1024] x [1024,1024]) via V_WMMA_F32_16X16X4_F32
// 18 x 64 output tiles, one wave per 16x16 tile, K loop of 256 WMMAs.
// ---------------------------------------------------------------------------
__global__ void gemm_T(const float* __restrict__ X, const float* __restrict__ W,
                       float* __restrict__ T) {
    int wid  = (blockIdx.x * blockDim.x + threadIdx.x) >> 5;   // 0..1151
    int lane = threadIdx.x & 31;
    int mt = wid >> 6;          // 0..17
    int nt = wid & 63;          // 0..63
    int row = mt * 16 + (lane & 15);
    int col = nt * 16 + (lane & 15);
    int khalf = (lane >> 4) * 2;          // lanes 0-15: K pair (0,1); 16-31: (2,3)
    const float* xrow = X + row * TWO_D;
    v8f c = {};
    for (int k0 = 0; k0 < TWO_D; k0 += 4) {
        int ka = k0 + khalf;
        v2f a, bf;
        a.x  = xrow[ka];
        a.y  = xrow[ka + 1];
        bf.x = W[ka * TWO_D + col];
        bf.y = W[(ka + 1) * TWO_D + col];
        c = __builtin_amdgcn_wmma_f32_16x16x4_f32(false, a, false, bf,
                                                  (short)0, c, false, false);
    }
    int mbase = mt * 16 + ((lane >> 4) << 3);   // D: M = r + 8*(lane/16)
    int nn = nt * 16 + (lane & 15);
#pragma unroll
    for (int r = 0; r < 8; ++r)
        T[(mbase + r) * TWO_D + nn] = c[r];
}

// ---------------------------------------------------------------------------
// Kernel D: Me[b] = T[b] * Y[b]^T  ([72,1024] x [1024,72]) via WMMA f32.
// 5x5 ragged tiles per batch; clamp loads, guard stores.
// ---------------------------------------------------------------------------
__global__ void gemm_Me(const float* __restrict__ T, const float* __restrict__ Y,
                        float* __restrict__ Me) {
    int wid = (blockIdx.x * blockDim.x + threadIdx.x) >> 5;
    if (wid >= Bq * 25) return;                 // uniform per wave
    int lane = threadIdx.x & 31;
    int b = wid / 25, t = wid % 25;
    int mt = t / 5, nt = t % 5;
    int m = mt * 16 + (lane & 15);
    int n = nt * 16 + (lane & 15);
    int mc = m < (Ee - 1) ? m : (Ee - 1);       // clamp ragged rows/cols
    int nc = n < (Ee - 1) ? n : (Ee - 1);
    const float* Tb = T + (b * Ee) * TWO_D + mc * TWO_D;
    const float* Yb = Y + (b * Ee) * TWO_D + nc * TWO_D;
    int khalf = (lane >> 4) * 2;
    v8f c = {};
    for (int k0 = 0; k0 < TWO_D; k0 += 4) {
        int ka = k0 + khalf;
        v2f a, bf;
        a.x  = Tb[ka];
        a.y  = Tb[ka + 1];
        bf.x = Yb[ka];                          // B[k][n] = Y[n][k]
        bf.y = Yb[ka + 1];
        c = __builtin_amdgcn_wmma_f32_16x16x4_f32(false, a, false, bf,
                                                  (short)0, c, false, false);
    }
    int mbase = mt * 16 + ((lane >> 4) << 3);
#pragma unroll
    for (int r = 0; r < 8; ++r) {
        int mm = mbase + r;
        if (mm < Ee && n < Ee)
            Me[b * Ee * Ee + mm * Ee + n] = c[r];   // Me[b,k,l], k=src edge, l=tgt edge
    }
}

// ---------------------------------------------------------------------------
// Kernel E: per-batch solver. Sparse-factored power iteration in LDS,
// Sinkhorn, softmax voting, displacement. One workgroup (576 thr) per batch.
// Deterministic: fixed-order loops + tree reductions, no atomics.
// ---------------------------------------------------------------------------
__global__ __launch_bounds__(RR) void solve(
        const float* __restrict__ U_src, const float* __restrict__ U_tgt,
        const float* __restrict__ G_src, const float* __restrict__ H_src,
        const float* __restrict__ G_tgt, const float* __restrict__ H_tgt,
        const float* __restrict__ P_src, const float* __restrict__ P_tgt,
        const float* __restrict__ Me_g, float* __restrict__ out) {
    __shared__ float sMe[Ee * Ee];       // Me[k*72+l]
    __shared__ float sT[Ee * Ee];        // t[l*72+k]
    __shared__ float sMp[RR];
    __shared__ float sV[RR];
    __shared__ float sWv[RR];
    __shared__ float sRed[RR];
    __shared__ int   eidx[4][Ee];        // gs, hs, gt, ht
    __shared__ int   cntGT[Nn], cntGS[Nn];
    __shared__ short listGT[Nn][Ee], listGS[Nn][Ee];
    __shared__ float sS[Nn * Nn];
    __shared__ float sSum[Nn], sMax[Nn];

    const int b   = blockIdx.x;
    const int tid = threadIdx.x;

    // --- per-edge node indices from one-hot incidence columns ---
    if (tid < 4 * Ee) {
        int which = tid / Ee, e = tid % Ee;
        const float* Ginc = (which == 0) ? G_src : (which == 1) ? H_src
                          : (which == 2) ? G_tgt : H_tgt;
        const float* col = Ginc + b * Nn * Ee + e;
        int idx = 0;
        for (int n = 0; n < Nn; ++n)
            if (col[n * Ee] > 0.5f) idx = n;
        eidx[which][e] = idx;
    }
    // --- Me tile into LDS ---
    for (int i = tid; i < Ee * Ee; i += RR) sMe[i] = Me_g[b * Ee * Ee + i];
    // --- Mp diagonal: mp[j*24+i] = dot(U_src[i], U_tgt[j]) ---
    {
        int i = tid % Nn, j = tid / Nn;
        const float* us = U_src + (b * Nn + i) * Dd;
        const float* ut = U_tgt + (b * Nn + j) * Dd;
        float acc = 0.f;
#pragma unroll 4
        for (int d = 0; d < Dd; ++d) acc += us[d] * ut[d];
        sMp[tid] = acc;
        sV[tid]  = 1.0f;
    }
    __syncthreads();

    // --- deterministic node -> edge lists (ascending edge order) ---
    if (tid < Nn) {                      // edges with gt[l] == node
        int c = 0;
        for (int l = 0; l < Ee; ++l)
            if (eidx[2][l] == tid) listGT[tid][c++] = (short)l;
        cntGT[tid] = c;
    } else if (tid < 2 * Nn) {           // edges with gs[k] == node
        int node = tid - Nn, c = 0;
        for (int k = 0; k < Ee; ++k)
            if (eidx[0][k] == node) listGS[node][c++] = (short)k;
        cntGS[node] = c;
    }
    __syncthreads();

    // --- power iteration: M*v via factored Kronecker/one-hot structure ---
    for (int it = 0; it < PI_ITERS; ++it) {
        // t[l,k] = Me[k,l] * v[ht[l]*24 + hs[k]]
        for (int p = tid; p < Ee * Ee; p += RR) {
            int l = p / Ee, k = p % Ee;
            sT[p] = sMe[k * Ee + l] * sV[eidx[3][l] * Nn + eidx[1][k]];
        }
        __syncthreads();
        // w[j*24+i] = mp*v + sum_{l:gt=j} sum_{k:gs=i} t[l,k]
        {
            int i = tid % Nn, j = tid / Nn;
            float acc = sMp[tid] * sV[tid];
            int ct = cntGT[j], cs = cntGS[i];
            for (int a = 0; a < ct; ++a) {
                const float* trow = sT + (int)listGT[j][a] * Ee;
                for (int c = 0; c < cs; ++c) acc += trow[listGS[i][c]];
            }
            sWv[tid]  = acc;
            sRed[tid] = acc * acc;
        }
        __syncthreads();
        // deterministic tree reduction of 576 squares
        if (tid < 64) sRed[tid] += sRed[512 + tid];
        __syncthreads();
        for (int s = 256; s >= 1; s >>= 1) {
            if (tid < s) sRed[tid] += sRed[tid + s];
            __syncthreads();
        }
        float norm = sqrtf(sRed[0]);
        sV[tid] = sWv[tid] / norm;
        __syncthreads();
    }

    // --- s[i,j] = v[j*24+i] + eps ---
    {
        int i = tid / Nn, j = tid % Nn;
        sS[tid] = sV[j * Nn + i] + 1e-10f;
    }
    __syncthreads();

    // --- Sinkhorn: even iters normalize columns (axis=1), odd rows ---
    for (int it = 0; it < SH_ITERS; ++it) {
        if (tid < Nn) {
            float ssum = 0.f;
            if ((it & 1) == 0)
                for (int i2 = 0; i2 < Nn; ++i2) ssum += sS[i2 * Nn + tid];
            else
                for (int j2 = 0; j2 < Nn; ++j2) ssum += sS[tid * Nn + j2];
            sSum[tid] = ssum;
        }
        __syncthreads();
        {
            int i = tid / Nn, j = tid % Nn;
            sS[tid] /= ((it & 1) == 0) ? sSum[j] : sSum[i];
        }
        __syncthreads();
    }

    // --- softmax(alpha * s) over rows ---
    if (tid < Nn) {
        float m = -3.0e38f;
        for (int j2 = 0; j2 < Nn; ++j2) m = fmaxf(m, sS[tid * Nn + j2]);
        sMax[tid] = m;
    }
    __syncthreads();
    sS[tid] = expf(200.0f * (sS[tid] - sMax[tid / Nn]));
    __syncthreads();
    if (tid < Nn) {
        float ssum = 0.f;
        for (int j2 = 0; j2 < Nn; ++j2) ssum += sS[tid * Nn + j2];
        sSum[tid] = ssum;
    }
    __syncthreads();
    sS[tid] /= sSum[tid / Nn];
    __syncthreads();

    // --- outputs: s [4,24,24] then d = s@P_tgt - P_src [4,24,2] ---
    out[b * RR + tid] = sS[tid];
    if (tid < Nn * 2) {
        int i = tid >> 1, k = tid & 1;
        float acc = -P_src[b * Nn * 2 + tid];
        for (int j = 0; j < Nn; ++j)
            acc += sS[i * Nn + j] * P_tgt[b * Nn * 2 + j * 2 + k];
        out[Bq * RR + b * Nn * 2 + tid] = acc;
    }
}

// ---------------------------------------------------------------------------
extern "C" void kernel_launch(void* const* d_in, const int* in_sizes, int n_in,
                              void* d_out, int out_size, void* d_ws, size_t ws_size,
                              hipStream_t stream) {
    const float* U_src   = (const float*)d_in[0];
    const float* U_tgt   = (const float*)d_in[1];
    const float* F_src   = (const float*)d_in[2];
    const float* F_tgt   = (const float*)d_in[3];
    const float* G_src   = (const float*)d_in[4];
    const float* G_tgt   = (const float*)d_in[5];
    const float* H_src   = (const float*)d_in[6];
    const float* H_tgt   = (const float*)d_in[7];
    // d_in[8..11]: ns_src, ns_tgt, K_G, K_H — unused (one-hot Kronecker structure exploited)
    const float* P_src   = (const float*)d_in[12];
    const float* P_tgt   = (const float*)d_in[13];
    const float* lambda1 = (const float*)d_in[14];
    const float* lambda2 = (const float*)d_in[15];
    float* out = (float*)d_out;

    float* Wm = (float*)d_ws;                     // 1024*1024
    float* X  = Wm + TWO_D * TWO_D;               // 288*1024
    float* Y  = X + NE * TWO_D;                   // 288*1024
    float* T  = Y + NE * TWO_D;                   // 288*1024
    float* Me = T + NE * TWO_D;                   // 4*72*72

    build_w  <<<(TWO_D * TWO_D) / 256, 256, 0, stream>>>(lambda1, lambda2, Wm);
    gather_xy<<<2 * NE, 256, 0, stream>>>(F_src, F_tgt, G_src, H_src, G_tgt, H_tgt, X, Y);
    gemm_T   <<<(18 * 64) / 8, 256, 0, stream>>>(X, Wm, T);         // 1152 waves
    gemm_Me  <<<13, 256, 0, stream>>>(T, Y, Me);                    // 100 waves (+guard)
    solve    <<<Bq, RR, 0, stream>>>(U_src, U_tgt, G_src, H_src, G_tgt, H_tgt,
                                     P_src, P_tgt, Me, out);
}